// OutputPPBlock_16286515986998
// MI455X (gfx1250) — compile-verified
//
#include <hip/hip_runtime.h>
#include <hip/hip_bf16.h>
#include <math.h>

// ---------------------------------------------------------------------------
// DimeNet-style OutputPPBlock on MI455X (gfx1250, wave32, WMMA)
//   E=800000 edges, HIDDEN=128, OUT_EMB=256, N_NODES=50000 (=3125*16)
// Stage 1: rbf-gate + scatter-add (HBM-bound; NT loads for streamed inputs,
//          fp32 atomics land in the 192MB L2 where h0 is resident)
// Stage 2: fp32 GEMM chain via V_WMMA_F32_16X16X4_F32, 32x64 wave tiles
// Stage 3: row-dot output projection
// ---------------------------------------------------------------------------

typedef __attribute__((ext_vector_type(2))) float v2f;
typedef __attribute__((ext_vector_type(4))) float v4f;
typedef __attribute__((ext_vector_type(8))) float v8f;

#define HIDDEN_DIM 128
#define OUT_EMB_DIM 256

// ---------------------------------------------------------------------------
// Kernel 1: g = (rbf @ W_rbf^T) * x ; atomic scatter-add into h0[node,128]
// One wave per edge; each lane owns 4 hidden channels. Streamed inputs use
// non-temporal loads to keep L2 free for h0 + GEMM weights.
// ---------------------------------------------------------------------------
__global__ void __launch_bounds__(256)
gate_scatter_kernel(const float* __restrict__ x,
                    const float* __restrict__ rbf,
                    const int*   __restrict__ idx,
                    const float* __restrict__ W_rbf,
                    float*       __restrict__ h0,
                    int E)
{
    const int e    = blockIdx.x * 8 + (threadIdx.x >> 5);
    const int lane = threadIdx.x & 31;
    if (e >= E) return;

    const float* rp = rbf + (size_t)e * 6;
    const float r0 = __builtin_nontemporal_load(rp + 0);
    const float r1 = __builtin_nontemporal_load(rp + 1);
    const float r2 = __builtin_nontemporal_load(rp + 2);
    const float r3 = __builtin_nontemporal_load(rp + 3);
    const float r4 = __builtin_nontemporal_load(rp + 4);
    const float r5 = __builtin_nontemporal_load(rp + 5);

    const v4f xv = __builtin_nontemporal_load(
        (const v4f*)(x + (size_t)e * HIDDEN_DIM + lane * 4));

    const int node = __builtin_nontemporal_load(idx + e);
    float* dst = h0 + (size_t)node * HIDDEN_DIM + lane * 4;

    #pragma unroll
    for (int j = 0; j < 4; ++j) {
        const float* wr = W_rbf + (size_t)(lane * 4 + j) * 6;
        const float g = r0 * wr[0] + r1 * wr[1] + r2 * wr[2]
                      + r3 * wr[3] + r4 * wr[4] + r5 * wr[5];
        atomicAdd(dst + j, g * xv[j]);   // global_atomic_add_f32, no return
    }
}

// ---------------------------------------------------------------------------
// Kernel 2: C[M,256] = act( A[M,K] @ W[256,K]^T + bias )
// Wave tile: 32 rows x 64 cols (2 A-frags x 4 B-frags -> 8 v8f accumulators).
// Block = 4 waves covering the full N=256; grid = ceil(M/32), tail guarded.
// Per K-step: 6 b64 loads feed 8 WMMAs (0.75 loads/wmma).
//
// Fragment layouts (ISA 7.12.2, fp32):
//   A 16x4 : lane L (L&15 = row) holds {K=0,K=1} if L<16 else {K=2,K=3}
//   B 4x16 : lane L (L&15 = col) holds {K=0,K=1} if L<16 else {K=2,K=3}
//   D 16x16: VGPR v, lane L -> row = v + 8*(L>=16), col = L&15
// ---------------------------------------------------------------------------
#define WMMA_F32(A_, B_, C_) \
    __builtin_amdgcn_wmma_f32_16x16x4_f32(false, (A_), false, (B_), (short)0, (C_), false, false)

template<int K, bool ACT>
__global__ void __launch_bounds__(128)
wmma_gemm_kernel(const float* __restrict__ A,
                 const float* __restrict__ W,
                 const float* __restrict__ bias,
                 float*       __restrict__ C,
                 int M)
{
    constexpr int N = OUT_EMB_DIM;
    const int lane = threadIdx.x & 31;
    const int wave = threadIdx.x >> 5;        // 0..3 -> 64-col slab
    const int m0   = blockIdx.x * 32;
    const int n0   = wave * 64;
    const int half = lane >> 4;               // K sub-pair select
    const int l15  = lane & 15;

    const int r0 = m0 + l15;
    const int r1 = m0 + 16 + l15;
    const int r1c = (r1 < M) ? r1 : (M - 1);  // clamp tail loads (stores guarded)

    const float* Ap0 = A + (size_t)r0  * K + half * 2;
    const float* Ap1 = A + (size_t)r1c * K + half * 2;
    const float* Wp0 = W + (size_t)(n0 +  0 + l15) * K + half * 2;
    const float* Wp1 = W + (size_t)(n0 + 16 + l15) * K + half * 2;
    const float* Wp2 = W + (size_t)(n0 + 32 + l15) * K + half * 2;
    const float* Wp3 = W + (size_t)(n0 + 48 + l15) * K + half * 2;

    v8f c00 = {}, c01 = {}, c02 = {}, c03 = {};
    v8f c10 = {}, c11 = {}, c12 = {}, c13 = {};

    #pragma unroll 4
    for (int k = 0; k < K; k += 4) {
        const v2f a0 = *(const v2f*)(Ap0 + k);
        const v2f a1 = *(const v2f*)(Ap1 + k);
        const v2f b0 = *(const v2f*)(Wp0 + k);
        const v2f b1 = *(const v2f*)(Wp1 + k);
        const v2f b2 = *(const v2f*)(Wp2 + k);
        const v2f b3 = *(const v2f*)(Wp3 + k);
        c00 = WMMA_F32(a0, b0, c00);
        c10 = WMMA_F32(a1, b0, c10);
        c01 = WMMA_F32(a0, b1, c01);
        c11 = WMMA_F32(a1, b1, c11);
        c02 = WMMA_F32(a0, b2, c02);
        c12 = WMMA_F32(a1, b2, c12);
        c03 = WMMA_F32(a0, b3, c03);
        c13 = WMMA_F32(a1, b3, c13);
    }

    v8f accs[2][4] = { { c00, c01, c02, c03 }, { c10, c11, c12, c13 } };

    #pragma unroll
    for (int mt = 0; mt < 2; ++mt) {
        const int rbase = m0 + mt * 16 + half * 8;
        #pragma unroll
        for (int t = 0; t < 4; ++t) {
            const int col = n0 + t * 16 + l15;
            const float bv = bias ? bias[col] : 0.0f;
            #pragma unroll
            for (int v = 0; v < 8; ++v) {
                const int row = rbase + v;
                if (row < M) {
                    float val = accs[mt][t][v] + bv;
                    if (ACT) val = val * (1.0f / (1.0f + __expf(-val)));  // silu
                    C[(size_t)row * N + col] = val;
                }
            }
        }
    }
}

// ---------------------------------------------------------------------------
// Kernel 3: out[m] = H[m,:] . W_out[0,:]   (N = 256, wave per row)
// ---------------------------------------------------------------------------
__global__ void __launch_bounds__(256)
rowdot_kernel(const float* __restrict__ H,
              const float* __restrict__ w,
              float*       __restrict__ out)
{
    const int row  = blockIdx.x * 8 + (threadIdx.x >> 5);
    const int lane = threadIdx.x & 31;
    const float* hr = H + (size_t)row * OUT_EMB_DIM + lane * 8;
    const float* wr = w + lane * 8;

    const float4 h0 = *(const float4*)(hr);
    const float4 h1 = *(const float4*)(hr + 4);
    const float4 w0 = *(const float4*)(wr);
    const float4 w1 = *(const float4*)(wr + 4);

    float s = h0.x * w0.x + h0.y * w0.y + h0.z * w0.z + h0.w * w0.w
            + h1.x * w1.x + h1.y * w1.y + h1.z * w1.z + h1.w * w1.w;

    #pragma unroll
    for (int off = 16; off > 0; off >>= 1)
        s += __shfl_xor(s, off, 32);

    if (lane == 0) out[row] = s;
}

// ---------------------------------------------------------------------------
extern "C" void kernel_launch(void* const* d_in, const int* in_sizes, int n_in,
                              void* d_out, int out_size, void* d_ws, size_t ws_size,
                              hipStream_t stream)
{
    const float* x      = (const float*)d_in[0];
    const float* rbf    = (const float*)d_in[1];
    const int*   idx    = (const int*)  d_in[2];
    // d_in[3] = num_nodes scalar (device); OUT=1 so out_size == num_nodes
    const float* W_rbf  = (const float*)d_in[4];
    const float* W_up   = (const float*)d_in[5];
    const float* W_lins = (const float*)d_in[6];
    const float* b_lins = (const float*)d_in[7];
    const float* W_out  = (const float*)d_in[8];
    float* out = (float*)d_out;

    const int E = in_sizes[0] / HIDDEN_DIM;
    const int M = out_size;                  // 50000

    // Workspace layout: h0[M,128] | h1[M,256] | h2[M,256]
    char* ws = (char*)d_ws;
    const size_t h0_bytes = (size_t)M * HIDDEN_DIM  * sizeof(float);
    const size_t h_bytes  = (size_t)M * OUT_EMB_DIM * sizeof(float);
    const size_t off1 = (h0_bytes + 255) & ~(size_t)255;
    const size_t off2 = off1 + ((h_bytes + 255) & ~(size_t)255);
    float* h0 = (float*)ws;
    float* h1 = (float*)(ws + off1);
    float* h2 = (float*)(ws + off2);

    // Zero scatter target every call (graph-capture legal).
    hipMemsetAsync(h0, 0, h0_bytes, stream);

    // Stage 1: gate + scatter-sum.
    gate_scatter_kernel<<<(E + 7) / 8, 256, 0, stream>>>(x, rbf, idx, W_rbf, h0, E);

    const int gblocks = (M + 31) / 32;       // 1563 (tail block guarded)

    // Stage 2: up-projection (K=128, no bias/act), then 3 silu layers (K=256).
    wmma_gemm_kernel<HIDDEN_DIM,  false><<<gblocks, 128, 0, stream>>>(h0, W_up, nullptr, h1, M);
    wmma_gemm_kernel<OUT_EMB_DIM, true ><<<gblocks, 128, 0, stream>>>(h1, W_lins + 0 * 65536, b_lins + 0 * 256, h2, M);
    wmma_gemm_kernel<OUT_EMB_DIM, true ><<<gblocks, 128, 0, stream>>>(h2, W_lins + 1 * 65536, b_lins + 1 * 256, h1, M);
    wmma_gemm_kernel<OUT_EMB_DIM, true ><<<gblocks, 128, 0, stream>>>(h1, W_lins + 2 * 65536, b_lins + 2 * 256, h2, M);

    // Stage 3: output projection.
    rowdot_kernel<<<M / 8, 256, 0, stream>>>(h2, W_out, out);
}